// DigitCaps_43903155699881
// MI455X (gfx1250) — compile-verified
//
#include <hip/hip_runtime.h>

typedef __attribute__((ext_vector_type(2))) float v2f;
typedef __attribute__((ext_vector_type(8))) float v8f;
typedef __attribute__((ext_vector_type(4))) int   i4v;

#define RTOT  1152
#define CCAP  10
#define DDIM  16
#define IDIM  8
#define BATCH 256
#define KTOT  (RTOT * IDIM)   // 9216
#define NTOT  (CCAP * DDIM)   // 160
#define NUM_IT 3

#define RB      8
#define BSTRIDE 12            // i-stride in Bs: float2-frag aligned, conflict-free
#define KSPLIT  8
#define RCHUNK  (RTOT / KSPLIT)   // 144 routes per block in split path
#define NST     (RCHUNK / RB)     // 18 stages

// Async global->LDS path (CDNA5): guarded so the file compiles on any toolchain.
#if defined(__has_builtin)
#if __has_builtin(__builtin_amdgcn_global_load_async_to_lds_b128)
#define HAVE_ASYNC_LDS 1
#endif
#endif

#ifdef HAVE_ASYNC_LDS
typedef __attribute__((address_space(1))) i4v as1_i4v;   // global int4
typedef __attribute__((address_space(3))) i4v as3_i4v;   // LDS int4
__device__ __forceinline__ void async_b128(const float* g, float* l) {
    void* gg = (void*)const_cast<float*>(g);
    void* ll = (void*)l;
    __builtin_amdgcn_global_load_async_to_lds_b128(
        (as1_i4v*)gg, (as3_i4v*)ll, 0, 0);
}
__device__ __forceinline__ void wait_async0() {
#if __has_builtin(__builtin_amdgcn_s_wait_asynccnt)
    __builtin_amdgcn_s_wait_asynccnt(0);
#else
    asm volatile("s_wait_asynccnt 0x0" ::: "memory");
#endif
}
#endif

// ws layout (floats): [0,11520) b_ij ; [11520,23040) c_ij ;
//                     [23040, 23040 + KSPLIT*40960) partial s (split path only)

__global__ void init_b_kernel(float* __restrict__ b_ij) {
    int t = blockIdx.x * blockDim.x + threadIdx.x;
    if (t < RTOT * CCAP) b_ij[t] = 0.0f;
}

__global__ void softmax_kernel(const float* __restrict__ b_ij,
                               float* __restrict__ c_ij) {
    int r = blockIdx.x * blockDim.x + threadIdx.x;
    if (r >= RTOT) return;
    float x[CCAP];
    float mx = -1e30f;
#pragma unroll
    for (int c = 0; c < CCAP; ++c) { x[c] = b_ij[r * CCAP + c]; mx = fmaxf(mx, x[c]); }
    float s = 0.0f;
#pragma unroll
    for (int c = 0; c < CCAP; ++c) { x[c] = __expf(x[c] - mx); s += x[c]; }
    float inv = 1.0f / s;
#pragma unroll
    for (int c = 0; c < CCAP; ++c) c_ij[r * CCAP + c] = x[c] * inv;
}

// ---------------------------------------------------------------------------
// Split-K partial GEMM: s_part[kz][b][c][d] over routes [kz*144,(kz+1)*144).
// Block = 4 waves, 64 batch rows, one capsule c, one K chunk.
// Double-buffered LDS stages; A staged via async global->LDS when available.
// ---------------------------------------------------------------------------
__global__ __launch_bounds__(128) void s_gemm_partial_kernel(
    const float* __restrict__ prim,   // [256,1152,8]
    const float* __restrict__ W,      // [1152,10,16,8]
    const float* __restrict__ c_ij,   // [1152,10]
    float* __restrict__ spart) {      // [KSPLIT,256,10,16]
    __shared__ __align__(16) float As[2][64 * 68];
    __shared__ __align__(16) float Bs[2][RB * DDIM * BSTRIDE];

    const int tid  = threadIdx.x;
    const int mb   = blockIdx.x;       // 0..3
    const int c    = blockIdx.y;       // 0..9
    const int kz   = blockIdx.z;       // 0..KSPLIT-1
    const int b0   = mb * 64;
    const int wv   = tid >> 5;
    const int l    = tid & 31;
    const int m    = l & 15;
    const int half = l >> 4;
    const int rbase = kz * RCHUNK;

    // Per-thread staging coordinates (constant across stages)
    const int brl0 = tid >> 5,        bw0 = tid & 31;          // B word 0
    const int brl1 = (128 + tid) >> 5, bw1 = (128 + tid) & 31; // B word 1
    const int bd0 = (bw0 * 4) >> 3, bi0 = (bw0 * 4) & 7;
    const int bd1 = (bw1 * 4) >> 3, bi1 = (bw1 * 4) & 7;

    v8f acc = {};

    // ---- prologue: stage 0 into buffer 0
    {
        const int r0 = rbase;
#ifdef HAVE_ASYNC_LDS
#pragma unroll
        for (int s = 0; s < 8; ++s) {
            int f = s * 128 + tid, row = f >> 4, c4 = f & 15;
            async_b128(prim + (size_t)(b0 + row) * KTOT + r0 * 8 + c4 * 4,
                       &As[0][row * 68 + c4 * 4]);
        }
#else
        float4 ar[8];
#pragma unroll
        for (int s = 0; s < 8; ++s) {
            int f = s * 128 + tid, row = f >> 4, c4 = f & 15;
            ar[s] = *(const float4*)(prim + (size_t)(b0 + row) * KTOT + r0 * 8 + c4 * 4);
        }
#pragma unroll
        for (int s = 0; s < 8; ++s) {
            int f = s * 128 + tid, row = f >> 4, c4 = f & 15;
            *(float4*)&As[0][row * 68 + c4 * 4] = ar[s];
        }
#endif
        float sc0 = c_ij[(r0 + brl0) * CCAP + c];
        float sc1 = c_ij[(r0 + brl1) * CCAP + c];
        float4 v0 = *(const float4*)(W + (size_t)((r0 + brl0) * CCAP + c) * 128 + bw0 * 4);
        float4 v1 = *(const float4*)(W + (size_t)((r0 + brl1) * CCAP + c) * 128 + bw1 * 4);
        v0.x *= sc0; v0.y *= sc0; v0.z *= sc0; v0.w *= sc0;
        v1.x *= sc1; v1.y *= sc1; v1.z *= sc1; v1.w *= sc1;
        *(float4*)&Bs[0][(brl0 * DDIM + bd0) * BSTRIDE + bi0] = v0;
        *(float4*)&Bs[0][(brl1 * DDIM + bd1) * BSTRIDE + bi1] = v1;
#ifdef HAVE_ASYNC_LDS
        wait_async0();
#endif
        __syncthreads();
    }

    const int arow = wv * 16 + m;
    for (int stg = 0; stg < NST; ++stg) {
        const int cur = stg & 1;
        const int nxt = cur ^ 1;
        const bool hn = (stg + 1 < NST);
        float4 v0, v1; float sc0 = 0.f, sc1 = 0.f;
        if (hn) {
            const int r0 = rbase + (stg + 1) * RB;
#ifdef HAVE_ASYNC_LDS
#pragma unroll
            for (int s = 0; s < 8; ++s) {
                int f = s * 128 + tid, row = f >> 4, c4 = f & 15;
                async_b128(prim + (size_t)(b0 + row) * KTOT + r0 * 8 + c4 * 4,
                           &As[nxt][row * 68 + c4 * 4]);
            }
#endif
            sc0 = c_ij[(r0 + brl0) * CCAP + c];
            sc1 = c_ij[(r0 + brl1) * CCAP + c];
            v0 = *(const float4*)(W + (size_t)((r0 + brl0) * CCAP + c) * 128 + bw0 * 4);
            v1 = *(const float4*)(W + (size_t)((r0 + brl1) * CCAP + c) * 128 + bw1 * 4);
        }
        // ---- compute on cur (16 WMMAs), overlapping in-flight loads
#pragma unroll
        for (int rl = 0; rl < RB; ++rl) {
#pragma unroll
            for (int ks = 0; ks < 8; ks += 4) {
                int kb = ks + half * 2;
                v2f a = *(const v2f*)&As[cur][arow * 68 + rl * 8 + kb];
                v2f b = *(const v2f*)&Bs[cur][(rl * DDIM + m) * BSTRIDE + kb];
                acc = __builtin_amdgcn_wmma_f32_16x16x4_f32(
                        false, a, false, b, (short)0, acc, false, false);
            }
        }
        if (hn) {
            v0.x *= sc0; v0.y *= sc0; v0.z *= sc0; v0.w *= sc0;
            v1.x *= sc1; v1.y *= sc1; v1.z *= sc1; v1.w *= sc1;
            *(float4*)&Bs[nxt][(brl0 * DDIM + bd0) * BSTRIDE + bi0] = v0;
            *(float4*)&Bs[nxt][(brl1 * DDIM + bd1) * BSTRIDE + bi1] = v1;
#ifndef HAVE_ASYNC_LDS
            // sync fallback for A: batched loads then stores
            const int r0 = rbase + (stg + 1) * RB;
            float4 ar[8];
#pragma unroll
            for (int s = 0; s < 8; ++s) {
                int f = s * 128 + tid, row = f >> 4, c4 = f & 15;
                ar[s] = *(const float4*)(prim + (size_t)(b0 + row) * KTOT + r0 * 8 + c4 * 4);
            }
#pragma unroll
            for (int s = 0; s < 8; ++s) {
                int f = s * 128 + tid, row = f >> 4, c4 = f & 15;
                *(float4*)&As[nxt][row * 68 + c4 * 4] = ar[s];
            }
#endif
        }
#ifdef HAVE_ASYNC_LDS
        wait_async0();
#endif
        __syncthreads();
    }

    // Write partial tile (no squash here): spart[kz][bg][c][d]
#pragma unroll
    for (int j = 0; j < 8; ++j) {
        int bg = b0 + wv * 16 + j + half * 8;
        spart[((size_t)kz * BATCH * CCAP + (size_t)bg * CCAP + c) * DDIM + m] = acc[j];
    }
}

// Deterministic fixed-order reduce over KSPLIT partials + squash. 1 thread per (b,c).
__global__ void squash_kernel(const float* __restrict__ sp, float* __restrict__ vout) {
    int t = blockIdx.x * blockDim.x + threadIdx.x;
    if (t >= BATCH * CCAP) return;
    float s[DDIM];
#pragma unroll
    for (int d = 0; d < DDIM; ++d) s[d] = 0.0f;
    for (int kz = 0; kz < KSPLIT; ++kz) {
        const float4* p = (const float4*)(sp + ((size_t)kz * BATCH * CCAP + t) * DDIM);
#pragma unroll
        for (int q = 0; q < 4; ++q) {
            float4 x = p[q];
            s[q * 4 + 0] += x.x; s[q * 4 + 1] += x.y;
            s[q * 4 + 2] += x.z; s[q * 4 + 3] += x.w;
        }
    }
    float n2 = 0.0f;
#pragma unroll
    for (int d = 0; d < DDIM; ++d) n2 += s[d] * s[d];
    float f = sqrtf(n2) / (1.0f + n2);
    float4* o = (float4*)(vout + (size_t)t * DDIM);
#pragma unroll
    for (int q = 0; q < 4; ++q) {
        float4 x; x.x = s[q*4+0]*f; x.y = s[q*4+1]*f; x.z = s[q*4+2]*f; x.w = s[q*4+3]*f;
        o[q] = x;
    }
}

// ---------------------------------------------------------------------------
// Fallback fused kernel (used when ws is too small for the split path).
// ---------------------------------------------------------------------------
__global__ __launch_bounds__(128) void s_gemm_squash_kernel(
    const float* __restrict__ prim, const float* __restrict__ W,
    const float* __restrict__ c_ij, float* __restrict__ vout) {
    __shared__ __align__(16) float As[64 * 68];
    __shared__ __align__(16) float Bs[RB * DDIM * BSTRIDE];
    const int tid = threadIdx.x, mb = blockIdx.x, c = blockIdx.y;
    const int b0 = mb * 64, wv = tid >> 5, l = tid & 31, m = l & 15, half = l >> 4;
    v8f acc = {};
    for (int r0 = 0; r0 < RTOT; r0 += RB) {
        __syncthreads();
#pragma unroll
        for (int s = 0; s < 8; ++s) {
            int f = s * 128 + tid, row = f >> 4, c4 = f & 15;
            float4 va = *(const float4*)(prim + (size_t)(b0 + row) * KTOT + r0 * 8 + c4 * 4);
            *(float4*)&As[row * 68 + c4 * 4] = va;
        }
#pragma unroll
        for (int s = 0; s < 2; ++s) {
            int f = s * 128 + tid, rl = f >> 5, w = f & 31;
            int elem = w * 4, d = elem >> 3, i0 = elem & 7;
            float scale = c_ij[(r0 + rl) * CCAP + c];
            float4 vb = *(const float4*)(W + (size_t)((r0 + rl) * CCAP + c) * 128 + elem);
            vb.x *= scale; vb.y *= scale; vb.z *= scale; vb.w *= scale;
            *(float4*)&Bs[(rl * DDIM + d) * BSTRIDE + i0] = vb;
        }
        __syncthreads();
        const int arow = wv * 16 + m;
#pragma unroll
        for (int rl = 0; rl < RB; ++rl) {
#pragma unroll
            for (int ks = 0; ks < 8; ks += 4) {
                int kb = ks + half * 2;
                v2f a = *(const v2f*)&As[arow * 68 + rl * 8 + kb];
                v2f b = *(const v2f*)&Bs[(rl * DDIM + m) * BSTRIDE + kb];
                acc = __builtin_amdgcn_wmma_f32_16x16x4_f32(
                        false, a, false, b, (short)0, acc, false, false);
            }
        }
    }
#pragma unroll
    for (int j = 0; j < 8; ++j) {
        float sv = acc[j];
        float n2 = sv * sv;
        n2 += __shfl_xor(n2, 1); n2 += __shfl_xor(n2, 2);
        n2 += __shfl_xor(n2, 4); n2 += __shfl_xor(n2, 8);
        float f = sqrtf(n2) / (1.0f + n2);
        int bg = b0 + wv * 16 + j + half * 8;
        vout[((size_t)bg * CCAP + c) * DDIM + m] = sv * f;
    }
}

// Agreement: b_ij[r,c] += (1/256)*sum_{i,d} W[r,c,d,i]*P[i,d], P = prim_rT @ V.
// One wave = two routes packed into M=16, one capsule; K = batch = 256.
__global__ __launch_bounds__(128) void agree_kernel(
    const float* __restrict__ prim, const float* __restrict__ W,
    const float* __restrict__ vbuf, float* __restrict__ b_ij) {
    const int tid = threadIdx.x;
    const int wid = blockIdx.x * 4 + (tid >> 5);
    const int rp = wid / CCAP, c = wid - rp * CCAP, r0 = rp * 2;
    const int l = tid & 31, m = l & 15, half = l >> 4;
    const int ra = r0 + (m >> 3), ia = m & 7;
    const size_t aoff = (size_t)ra * IDIM + ia;
    const size_t boff = (size_t)c * DDIM + m;
    v8f acc = {};
    for (int k0 = 0; k0 < BATCH; k0 += 4) {
        int kb = k0 + half * 2;
        v2f a, b;
        a.x = prim[(size_t)kb * KTOT + aoff];
        a.y = prim[(size_t)(kb + 1) * KTOT + aoff];
        b.x = vbuf[(size_t)kb * NTOT + boff];
        b.y = vbuf[(size_t)(kb + 1) * NTOT + boff];
        acc = __builtin_amdgcn_wmma_f32_16x16x4_f32(
                false, a, false, b, (short)0, acc, false, false);
    }
    const int r = (half == 0) ? r0 : (r0 + 1);
    const float* wp = W + ((size_t)(r * CCAP + c) * DDIM + m) * IDIM;
    float4 w0 = *(const float4*)(wp);
    float4 w1 = *(const float4*)(wp + 4);
    float p = acc[0]*w0.x + acc[1]*w0.y + acc[2]*w0.z + acc[3]*w0.w
            + acc[4]*w1.x + acc[5]*w1.y + acc[6]*w1.z + acc[7]*w1.w;
    p += __shfl_xor(p, 1); p += __shfl_xor(p, 2);
    p += __shfl_xor(p, 4); p += __shfl_xor(p, 8);
    if (m == 0) b_ij[r * CCAP + c] += p * (1.0f / 256.0f);
}

extern "C" void kernel_launch(void* const* d_in, const int* in_sizes, int n_in,
                              void* d_out, int out_size, void* d_ws, size_t ws_size,
                              hipStream_t stream) {
    (void)in_sizes; (void)n_in; (void)out_size;
    const float* prim = (const float*)d_in[0];   // [256,1152,8]
    const float* W    = (const float*)d_in[1];   // [1,1152,10,16,8]
    float* out  = (float*)d_out;                 // [256,10,16,1] -> holds v
    float* b_ij = (float*)d_ws;
    float* c_ij = b_ij + RTOT * CCAP;
    float* spart = c_ij + RTOT * CCAP;

    const size_t need = (size_t)(2 * RTOT * CCAP + KSPLIT * BATCH * CCAP * DDIM) * sizeof(float);
    const bool split = (ws_size >= need);

    init_b_kernel<<<(RTOT * CCAP + 127) / 128, 128, 0, stream>>>(b_ij);

    for (int it = 0; it < NUM_IT; ++it) {
        softmax_kernel<<<(RTOT + 127) / 128, 128, 0, stream>>>(b_ij, c_ij);
        if (split) {
            dim3 grid(4, CCAP, KSPLIT);
            s_gemm_partial_kernel<<<grid, 128, 0, stream>>>(prim, W, c_ij, spart);
            squash_kernel<<<(BATCH * CCAP + 127) / 128, 128, 0, stream>>>(spart, out);
        } else {
            dim3 grid(4, CCAP);
            s_gemm_squash_kernel<<<grid, 128, 0, stream>>>(prim, W, c_ij, out);
        }
        if (it + 1 < NUM_IT) {
            agree_kernel<<<(RTOT / 2) * CCAP / 4, 128, 0, stream>>>(prim, W, out, b_ij);
        }
    }
}